// SingleRoIExtractor_17600775979252
// MI455X (gfx1250) — compile-verified
//
#include <hip/hip_runtime.h>

typedef __attribute__((ext_vector_type(2))) float v2f;
typedef __attribute__((ext_vector_type(8))) float v8f;

#define OUTSZ  7
#define S14    14          // out_size * sample_num
#define NCH    256
#define LDSTR  272         // 256 + 16 pad: bank-conflict-free B-tile reads

__global__ __launch_bounds__(256)
void roi_extract_wmma(const float* __restrict__ f0,
                      const float* __restrict__ f1,
                      const float* __restrict__ f2,
                      const float* __restrict__ f3,
                      const float* __restrict__ rois,
                      float* __restrict__ out)
{
    __shared__ float V[2 * S14 * LDSTR];   // 28 samples x 256 ch (+pad) ~30.5 KB

    const int roi = blockIdx.x;
    const int tid = threadIdx.x;           // tid == channel for the gather stage

    // ---------------- per-ROI setup (uniform across block) ----------------
    const float bF = rois[roi * 5 + 0];
    const float x1 = rois[roi * 5 + 1];
    const float y1 = rois[roi * 5 + 2];
    const float x2 = rois[roi * 5 + 3];
    const float y2 = rois[roi * 5 + 4];

    const float scale = sqrtf((x2 - x1 + 1.0f) * (y2 - y1 + 1.0f));
    int lvl = (int)floorf(log2f(scale * (1.0f / 56.0f) + 1e-6f));
    lvl = lvl < 0 ? 0 : (lvl > 3 ? 3 : lvl);

    const int H = (lvl == 0) ? 200 : (lvl == 1) ? 100 : (lvl == 2) ? 50 : 25;
    const int W = (lvl == 0) ? 304 : (lvl == 1) ? 152 : (lvl == 2) ? 76 : 38;
    const float* __restrict__ f =
        (lvl == 0) ? f0 : (lvl == 1) ? f1 : (lvl == 2) ? f2 : f3;
    const float ss = 1.0f / (float)(4 << lvl);

    const int   b   = (int)bF;
    const float x1s = x1 * ss;
    const float y1s = y1 * ss;
    const float rw  = fmaxf(x2 * ss - x1s, 1.0f);
    const float rh  = fmaxf(y2 * ss - y1s, 1.0f);
    const float bw  = rw * (1.0f / (float)S14);
    const float bh  = rh * (1.0f / (float)S14);

    const size_t chanBase = ((size_t)b * NCH + (size_t)tid) * (size_t)(H * W);

    const int lane   = tid & 31;
    const int waveId = tid >> 5;
    const int half   = lane >> 4;   // 0: lanes 0-15, 1: lanes 16-31
    const int lm     = lane & 15;

    // --- A-matrix (pooling weights), invariant over py and channel tile ---
    // A layout (16x4 f32): lane L holds M=L%16; VGPR0 = K(k0+2*half), VGPR1 = K+1.
    v2f aop[7];
    #pragma unroll
    for (int i = 0; i < 7; ++i) {
        const int kk  = 4 * i + 2 * half;
        const int sx0 = kk % S14;
        const int sx1 = (kk + 1) % S14;
        aop[i].x = (lm < OUTSZ && (sx0 >> 1) == lm) ? 0.25f : 0.0f;
        aop[i].y = (lm < OUTSZ && (sx1 >> 1) == lm) ? 0.25f : 0.0f;
    }

    for (int py = 0; py < OUTSZ; ++py) {
        // ---------- stage 1: bilinear gather of 28 samples for channel=tid ----------
        for (int s = 0; s < 2 * S14; ++s) {
            const int   syi = 2 * py + (s >= S14 ? 1 : 0);
            const int   sxi = (s >= S14) ? (s - S14) : s;
            const float ysf = y1s + ((float)syi + 0.5f) * bh;
            const float xsf = x1s + ((float)sxi + 0.5f) * bw;

            float val = 0.0f;
            const bool vy = (ysf >= -1.0f) && (ysf <= (float)H);
            const bool vx = (xsf >= -1.0f) && (xsf <= (float)W);
            if (vy && vx) {
                float y = fmaxf(ysf, 0.0f);
                float x = fmaxf(xsf, 0.0f);
                int yl = (int)floorf(y);
                int xl = (int)floorf(x);
                if (yl >= H - 1) { y = (float)(H - 1); yl = H - 1; }
                if (xl >= W - 1) { x = (float)(W - 1); xl = W - 1; }
                const int yh = (yl + 1 < H - 1) ? (yl + 1) : (H - 1);
                const int xh = (xl + 1 < W - 1) ? (xl + 1) : (W - 1);
                const float ly = y - (float)yl;
                const float lx = x - (float)xl;
                const float hy = 1.0f - ly;
                const float hx = 1.0f - lx;
                const float* __restrict__ rl  = f + chanBase + (size_t)yl * (size_t)W;
                const float* __restrict__ rh2 = f + chanBase + (size_t)yh * (size_t)W;
                val = hy * hx * rl[xl] + hy * lx * rl[xh]
                    + ly * hx * rh2[xl] + ly * lx * rh2[xh];
            }
            V[s * LDSTR + tid] = val;
        }
        __syncthreads();

        // ---------- stage 2: 2x2 average pool as WMMA f32 16x16x4 ----------
        // out[16(px) x 16(ch)] = P[16x28] * V[28x16],  K in steps of 4.
        for (int t = 0; t < 2; ++t) {
            const int n0 = (waveId * 2 + t) * 16;     // channel tile base
            v8f acc = {0.f, 0.f, 0.f, 0.f, 0.f, 0.f, 0.f, 0.f};
            #pragma unroll
            for (int i = 0; i < 7; ++i) {
                const int kk = 4 * i + 2 * half;      // B rows striped like A
                v2f bm;
                bm.x = V[kk * LDSTR + n0 + lm];
                bm.y = V[(kk + 1) * LDSTR + n0 + lm];
                acc = __builtin_amdgcn_wmma_f32_16x16x4_f32(
                          false, aop[i], false, bm, (short)0, acc, false, false);
            }
            // D layout: VGPR r -> M = r + 8*half, N = lm.
            // Valid output rows (px<7) exist only in the half==0 lanes, r=0..6,
            // and land at 7 consecutive addresses -> single divergent region,
            // non-temporal stores (write-once output must not evict the
            // L2-resident feature pyramid).
            if (half == 0) {
                const int ch = n0 + lm;
                float* __restrict__ p =
                    out + ((size_t)roi * NCH + (size_t)ch) * (OUTSZ * OUTSZ)
                        + (size_t)py * OUTSZ;
                #pragma unroll
                for (int r = 0; r < OUTSZ; ++r) {
                    __builtin_nontemporal_store(acc[r], p + r);
                }
            }
            __syncthreads();   // also isolates LDS reuse across py iterations
        }
    }
}

extern "C" void kernel_launch(void* const* d_in, const int* in_sizes, int n_in,
                              void* d_out, int out_size, void* d_ws, size_t ws_size,
                              hipStream_t stream) {
    const float* f0   = (const float*)d_in[0];
    const float* f1   = (const float*)d_in[1];
    const float* f2   = (const float*)d_in[2];
    const float* f3   = (const float*)d_in[3];
    const float* rois = (const float*)d_in[4];
    float* out = (float*)d_out;

    const int nroi = in_sizes[4] / 5;   // rois are [K,5]
    roi_extract_wmma<<<nroi, 256, 0, stream>>>(f0, f1, f2, f3, rois, out);
}